// CorModule_14628658610929
// MI455X (gfx1250) — compile-verified
//
#include <hip/hip_runtime.h>
#include <hip/hip_bf16.h>
#include <math.h>

// ---------------------------------------------------------------------------
// C = L * L^T where L is the 4096x4096 lower-triangular "Cholesky factor of a
// correlation matrix" built from tanh-transformed unconstrained params.
//
// Kernel 1: one 256-thread block per row; per-thread 16-wide cumprod segment,
//           LDS Hillis-Steele scan for the exclusive prefix product, then
//           L[i,j] = tanh(p)*sqrt(s_j); diag sqrt(s_i); upper zero.
// Kernel 2: 128x128 C-tile per block, K staged 32-wide in LDS, fp32 WMMA
//           (v_wmma_f32_16x16x4_f32). B = L^T means the ISA B-fragment layout
//           equals the A-fragment layout of L, so one LDS read pattern feeds
//           both operands. K loop truncated by triangularity.
// ---------------------------------------------------------------------------

typedef __attribute__((ext_vector_type(2))) float v2f;
typedef __attribute__((ext_vector_type(8))) float v8f;

#define N 4096
#define BT 128        // C tile (BT x BT) per block
#define KB 32         // K slab staged in LDS
#define LDT 36        // LDS row stride in floats (even, 16B aligned, no bank conflicts)

// ------------------------------ Kernel 1 -----------------------------------
__global__ __launch_bounds__(256) void build_chol_kernel(
    const float* __restrict__ p, float* __restrict__ L) {
  const int i = blockIdx.x;                 // row
  const int t = threadIdx.x;                // 0..255
  const int off = (i * (i - 1)) >> 1;       // fits in int32 for N=4096

  __shared__ float sprod[256];

  float z[16];
  float lp = 1.0f;
  const int j0 = t * 16;
#pragma unroll
  for (int u = 0; u < 16; ++u) {
    const int j = j0 + u;
    float zz = 0.0f;
    if (j < i) {
      zz = tanhf(p[off + j]);
      lp *= (1.0f - zz * zz);
    }
    z[u] = zz;
  }
  sprod[t] = lp;
  __syncthreads();

  // Inclusive Hillis-Steele scan (product) over the 256 per-thread products.
#pragma unroll
  for (int s = 1; s < 256; s <<= 1) {
    const float mine = sprod[t];
    const float prev = (t >= s) ? sprod[t - s] : 1.0f;
    __syncthreads();
    sprod[t] = mine * prev;
    __syncthreads();
  }

  float s = (t == 0) ? 1.0f : sprod[t - 1];  // exclusive prefix product
  float* row = L + (size_t)i * N;
#pragma unroll
  for (int u = 0; u < 16; ++u) {
    const int j = j0 + u;
    float out;
    if (j < i) {
      out = z[u] * sqrtf(s);
      s *= (1.0f - z[u] * z[u]);
    } else if (j == i) {
      out = sqrtf(s);
    } else {
      out = 0.0f;
    }
    row[j] = out;
  }
}

// ------------------------------ Kernel 2 -----------------------------------
__global__ __launch_bounds__(256) void syrk_wmma_kernel(
    const float* __restrict__ L, float* __restrict__ C) {
  __shared__ float As[BT * LDT];
  __shared__ float Bs[BT * LDT];

  const int bm = blockIdx.y;
  const int bn = blockIdx.x;
  const int tid  = threadIdx.x;
  const int lane = tid & 31;
  const int wave = tid >> 5;      // 0..7
  const int wr   = wave & 3;      // wave row group: 32 rows of C
  const int wc   = wave >> 2;     // wave col group: 64 cols of C
  const int hi   = lane >> 4;     // 0/1 half-wave
  const int l15  = lane & 15;

  v8f acc[2][4];
#pragma unroll
  for (int r = 0; r < 2; ++r)
#pragma unroll
    for (int c = 0; c < 4; ++c) acc[r][c] = (v8f){};

  // Triangularity: C[i][j] needs k <= min(i,j) only.
  const int mn   = (bm < bn) ? bm : bn;
  const int kEnd = (mn + 1) * BT;

  // Global -> LDS staging indices: 8 threads x 16B per row, 32 rows per pass.
  const int lrow = tid >> 3;           // 0..31
  const int lcol = (tid & 7) * 4;      // 0,4,...,28
  const float* Abase = L + (size_t)(bm * BT) * N;
  const float* Bbase = L + (size_t)(bn * BT) * N;

  for (int k0 = 0; k0 < kEnd; k0 += KB) {
    __syncthreads();  // previous slab's LDS reads done
#pragma unroll
    for (int rr = 0; rr < BT; rr += 32) {
      const int r = rr + lrow;
      const float4 a = *(const float4*)(Abase + (size_t)r * N + k0 + lcol);
      *(float4*)(&As[r * LDT + lcol]) = a;
      const float4 b = *(const float4*)(Bbase + (size_t)r * N + k0 + lcol);
      *(float4*)(&Bs[r * LDT + lcol]) = b;
    }
    __syncthreads();

#pragma unroll
    for (int kk = 0; kk < KB; kk += 4) {
      // fp32 WMMA A-fragment: lane<16 -> K=kk,kk+1 ; lane>=16 -> K=kk+2,kk+3,
      // row M = l15. Since B = L^T, the B-fragment of column-block n is the
      // A-fragment of L's row-block n: identical read pattern.
      v2f af[2], bf[4];
#pragma unroll
      for (int r = 0; r < 2; ++r)
        af[r] = *(const v2f*)(&As[(wr * 32 + r * 16 + l15) * LDT + kk + 2 * hi]);
#pragma unroll
      for (int c = 0; c < 4; ++c)
        bf[c] = *(const v2f*)(&Bs[(wc * 64 + c * 16 + l15) * LDT + kk + 2 * hi]);

#pragma unroll
      for (int r = 0; r < 2; ++r)
#pragma unroll
        for (int c = 0; c < 4; ++c)
          acc[r][c] = __builtin_amdgcn_wmma_f32_16x16x4_f32(
              /*neg_a=*/false, af[r], /*neg_b=*/false, bf[c],
              /*c_mod=*/(short)0, acc[r][c],
              /*reuse_a=*/false, /*reuse_b=*/false);
    }
  }

  // Store: C/D layout — VGPR v: lanes 0-15 hold M=v, lanes 16-31 hold M=v+8.
#pragma unroll
  for (int r = 0; r < 2; ++r) {
    const int mbase = bm * BT + wr * 32 + r * 16 + hi * 8;
#pragma unroll
    for (int c = 0; c < 4; ++c) {
      const int nn = bn * BT + wc * 64 + c * 16 + l15;
#pragma unroll
      for (int v = 0; v < 8; ++v) {
        C[(size_t)(mbase + v) * N + nn] = acc[r][c][v];
      }
    }
  }
}

// ------------------------------ Launch -------------------------------------
extern "C" void kernel_launch(void* const* d_in, const int* in_sizes, int n_in,
                              void* d_out, int out_size, void* d_ws, size_t ws_size,
                              hipStream_t stream) {
  const float* params = (const float*)d_in[0];   // N*(N-1)/2 floats
  float* C = (float*)d_out;                      // N*N floats
  float* L = (float*)d_ws;                       // N*N floats scratch (64 MB)

  (void)in_sizes; (void)n_in; (void)out_size; (void)ws_size;

  build_chol_kernel<<<N, 256, 0, stream>>>(params, L);

  dim3 grid(N / BT, N / BT);  // 32 x 32
  syrk_wmma_kernel<<<grid, 256, 0, stream>>>(L, C);
}